// MultiHeadSelfAttention_30537217474867
// MI455X (gfx1250) — compile-verified
//
#include <hip/hip_runtime.h>
#include <hip/hip_bf16.h>
#include <math.h>

// ---------------------------------------------------------------------------
// Relative-position multi-head attention for MI455X (gfx1250), f32 WMMA path.
//   B=8, S=1024, D=512, H=8, dh=64, P=2*128+1=257
// Pipeline:
//   K1: QKV = x @ W_in + b_in          -> Q,K,V in [B,H,S,dh] (ws)
//   K2: per (b,h,q-block16) wave:
//         Qp  = Qtile @ posK^T (WMMA, LDS)
//         S   = (Q K^T + gather(Qp)) * 1/8, two-pass softmax
//         O   = (P @ V + Ap @ posV) / rowsum       (all WMMA f32 16x16x4)
//   K3: out = O @ W_out + b_out
// ---------------------------------------------------------------------------

namespace {

constexpr int Bn   = 8;
constexpr int Sn   = 1024;
constexpr int Dn   = 512;
constexpr int Hn   = 8;
constexpr int DHn  = 64;
constexpr int Pn   = 257;    // 2*MAX_POS + 1
constexpr int PPAD = 272;    // 17 tiles of 16
constexpr int MAXP = 128;

typedef __attribute__((ext_vector_type(2))) float v2f;
typedef __attribute__((ext_vector_type(8))) float v8f;

__device__ __forceinline__ v8f wmma4(v2f a, v2f b, v8f c) {
  // V_WMMA_F32_16X16X4_F32 : D = A(16x4) * B(4x16) + C(16x16), all f32
  return __builtin_amdgcn_wmma_f32_16x16x4_f32(false, a, false, b, (short)0, c,
                                               false, false);
}

__device__ __forceinline__ float redmax16(float v) {
#pragma unroll
  for (int off = 1; off < 16; off <<= 1) v = fmaxf(v, __shfl_xor(v, off, 32));
  return v;
}
__device__ __forceinline__ float redsum16(float v) {
#pragma unroll
  for (int off = 1; off < 16; off <<= 1) v += __shfl_xor(v, off, 32);
  return v;
}

// ---------------- K1: QKV projection, scatter to head layout ----------------
// waves = (8192/16 Mtiles) * (1536/64 Nstrips) = 512*24; 8 waves / block.
__global__ __launch_bounds__(256) void qkv_gemm(const float* __restrict__ x,
                                                const float* __restrict__ Win,
                                                const float* __restrict__ bin,
                                                float* __restrict__ Q,
                                                float* __restrict__ K,
                                                float* __restrict__ V) {
  const int lane = threadIdx.x & 31;
  const int wid  = (blockIdx.x * blockDim.x + threadIdx.x) >> 5;
  const int mt   = wid / 24;           // 0..511  (rows of x, 16 each)
  const int ns   = wid % 24;           // 0..23   (64-col strips of W_in)
  const int m    = lane & 15;
  const int hl   = lane >> 4;
  const int col0 = ns * 64;
  const float* arow = x + (size_t)(mt * 16 + m) * Dn;

  v8f acc[4] = {v8f{}, v8f{}, v8f{}, v8f{}};
#pragma unroll 4
  for (int ks = 0; ks < Dn / 4; ++ks) {
    const int kb = ks * 4 + hl * 2;
    v2f a = *(const v2f*)(arow + kb);
    const float* w0 = Win + (size_t)kb * (3 * Dn) + col0 + m;
    const float* w1 = w0 + 3 * Dn;
#pragma unroll
    for (int j = 0; j < 4; ++j) {
      v2f b;
      b.x = w0[16 * j];
      b.y = w1[16 * j];
      acc[j] = wmma4(a, b, acc[j]);
    }
  }

  const int which = ns >> 3;           // 0:Q 1:K 2:V
  const int h     = ns & 7;
  float* dst = (which == 0) ? Q : (which == 1) ? K : V;
#pragma unroll
  for (int j = 0; j < 4; ++j) {
    const int   d    = j * 16 + m;
    const float bias = bin[col0 + d];
#pragma unroll
    for (int r = 0; r < 8; ++r) {
      const int grow = mt * 16 + r + hl * 8;   // global row = b*S + s
      const int bb = grow >> 10, ss = grow & (Sn - 1);
      dst[((size_t)(bb * Hn + h) * Sn + ss) * DHn + d] = acc[j][r] + bias;
    }
  }
}

// ---------------- K2: attention, one wave per (b,h,q-block16) ---------------
__global__ __launch_bounds__(32) void attn(const float* __restrict__ Q,
                                           const float* __restrict__ K,
                                           const float* __restrict__ V,
                                           const int* __restrict__ mask,
                                           const float* __restrict__ posK,
                                           const float* __restrict__ posV,
                                           float* __restrict__ O) {
  __shared__ float Qp[16][PPAD];   // Qtile @ posK^T
  __shared__ float Ap[16][PPAD];   // bucketed probabilities
  __shared__ float Pt[16][16];     // C-layout -> A-layout staging

  const int lane = threadIdx.x;
  const int m  = lane & 15;
  const int hl = lane >> 4;
  const int qb = blockIdx.x & (Sn / 16 - 1);
  const int bh = blockIdx.x >> 6;
  const int h  = bh & (Hn - 1);
  const int b  = bh >> 3;
  const int q0 = qb * 16;
  const float* Qh = Q + (size_t)bh * Sn * DHn;
  const float* Kh = K + (size_t)bh * Sn * DHn;
  const float* Vh = V + (size_t)bh * Sn * DHn;
  const int*   mr = mask + b * Sn;

  // Q tile as A-fragments (row m, k-dims striped per WMMA A layout)
  v2f qa[16];
#pragma unroll
  for (int ks = 0; ks < 16; ++ks)
    qa[ks] = *(const v2f*)(Qh + (size_t)(q0 + m) * DHn + ks * 4 + hl * 2);

  for (int i = lane; i < 16 * PPAD; i += 32) (&Ap[0][0])[i] = 0.0f;

  // ---- Qp = Qtile(16x64) @ posK^T(64x257), padded to 272 cols ----
  for (int j = 0; j < PPAD / 16; ++j) {
    const int  p   = j * 16 + m;
    const bool pin = (p < Pn);
    const float* pk = posK + (size_t)p * DHn + hl * 2;
    v8f acc = {};
#pragma unroll
    for (int ks = 0; ks < 16; ++ks) {
      float bx = 0.f, by = 0.f;
      if (pin) { v2f t = *(const v2f*)(pk + ks * 4); bx = t.x; by = t.y; }
      v2f bf; bf.x = bx; bf.y = by;
      acc = wmma4(qa[ks], bf, acc);
    }
#pragma unroll
    for (int r = 0; r < 8; ++r) Qp[r + hl * 8][p] = acc[r];
  }

  // ---- Pass 1: row maxima over all k ----
  float rmax[8];
#pragma unroll
  for (int r = 0; r < 8; ++r) rmax[r] = -3.4e38f;

  for (int kt = 0; kt < Sn / 16; ++kt) {
    const int k0 = kt * 16;
    const float* kp = Kh + (size_t)(k0 + m) * DHn + hl * 2;
    if (kt + 1 < Sn / 16)
      __builtin_prefetch((const void*)(kp + 16 * DHn), 0, 1);
    v8f acc = {};
#pragma unroll
    for (int ks = 0; ks < 16; ++ks)
      acc = wmma4(qa[ks], *(const v2f*)(kp + ks * 4), acc);
    const int  kk  = k0 + m;
    const bool mok = (mr[kk] != 0);
#pragma unroll
    for (int r = 0; r < 8; ++r) {
      const int row  = r + hl * 8;
      int rr = kk - (q0 + row);
      rr = rr < -MAXP ? -MAXP : (rr > MAXP ? MAXP : rr);
      float s = (acc[r] + Qp[row][rr + MAXP]) * 0.125f;
      s = mok ? s : -3.4e38f;
      rmax[r] = fmaxf(rmax[r], s);
    }
  }
#pragma unroll
  for (int r = 0; r < 8; ++r) rmax[r] = redmax16(rmax[r]);

  // ---- Pass 2: exp, row sums, O1 += P@V, bucket Ap ----
  float rsum[8];
#pragma unroll
  for (int r = 0; r < 8; ++r) rsum[r] = 0.0f;
  v8f oacc[4] = {v8f{}, v8f{}, v8f{}, v8f{}};

  for (int kt = 0; kt < Sn / 16; ++kt) {
    const int k0 = kt * 16;
    const float* kp = Kh + (size_t)(k0 + m) * DHn + hl * 2;
    if (kt + 1 < Sn / 16)
      __builtin_prefetch((const void*)(Vh + (size_t)(k0 + 16 + m) * DHn), 0, 1);
    v8f acc = {};
#pragma unroll
    for (int ks = 0; ks < 16; ++ks)
      acc = wmma4(qa[ks], *(const v2f*)(kp + ks * 4), acc);

    const int  kk  = k0 + m;
    const bool mok = (mr[kk] != 0);
    const int  dt  = k0 - q0;
    const bool anyclip = (dt <= -113) || (dt >= 113);
#pragma unroll
    for (int r = 0; r < 8; ++r) {
      const int row = r + hl * 8;
      const int rr  = kk - (q0 + row);
      const int rc  = rr < -MAXP ? -MAXP : (rr > MAXP ? MAXP : rr);
      float s = (acc[r] + Qp[row][rc + MAXP]) * 0.125f;
      float e = mok ? __expf(s - rmax[r]) : 0.0f;
      rsum[r] += e;
      Pt[row][m] = e;
      if (anyclip) {
        // band positions are unique (one k per p) -> plain store;
        // clipped tails are shfl-reduced, added by one lane (deterministic).
        if (rr > -MAXP && rr < MAXP) Ap[row][rr + MAXP] = e;
        float slo = redsum16(rr <= -MAXP ? e : 0.0f);
        float shi = redsum16(rr >=  MAXP ? e : 0.0f);
        if (m == 0) {
          Ap[row][0]        += slo;
          Ap[row][2 * MAXP] += shi;
        }
      } else {
        Ap[row][rr + MAXP] = e;
      }
    }

    // O1 += Ptile(16x16) @ Vtile(16x64)
#pragma unroll
    for (int ks = 0; ks < 4; ++ks) {
      v2f a = *(const v2f*)(&Pt[m][ks * 4 + hl * 2]);
      const float* vp = Vh + (size_t)(k0 + ks * 4 + hl * 2) * DHn;
#pragma unroll
      for (int j = 0; j < 4; ++j) {
        v2f bf;
        bf.x = vp[j * 16 + m];
        bf.y = vp[DHn + j * 16 + m];
        oacc[j] = wmma4(a, bf, oacc[j]);
      }
    }
  }
#pragma unroll
  for (int r = 0; r < 8; ++r) rsum[r] = redsum16(rsum[r]);

  // ---- O2 += Ap(16x272) @ posV(272x64)  (posV rows >=257 read as zero) ----
  for (int jt = 0; jt < PPAD / 16; ++jt) {
#pragma unroll
    for (int ks = 0; ks < 4; ++ks) {
      const int pb = jt * 16 + ks * 4 + hl * 2;
      v2f a = *(const v2f*)(&Ap[m][pb]);
      v2f bf;
      bf.x = (pb     < Pn) ? posV[(size_t)pb * DHn] : 0.0f;        // patched per j below
      (void)bf;
#pragma unroll
      for (int j = 0; j < 4; ++j) {
        const int d = j * 16 + m;
        v2f bb;
        bb.x = (pb     < Pn) ? posV[(size_t)(pb)     * DHn + d] : 0.0f;
        bb.y = (pb + 1 < Pn) ? posV[(size_t)(pb + 1) * DHn + d] : 0.0f;
        oacc[j] = wmma4(a, bb, oacc[j]);
      }
    }
  }

  // ---- normalize + store to attn-out [B,S,D] (head-interleaved cols) ----
#pragma unroll
  for (int j = 0; j < 4; ++j) {
#pragma unroll
    for (int r = 0; r < 8; ++r) {
      const int qrow = q0 + r + hl * 8;
      O[((size_t)b * Sn + qrow) * Dn + h * DHn + j * 16 + m] =
          oacc[j][r] * (1.0f / rsum[r]);
    }
  }
}

// ---------------- K3: output projection ----------------
// waves = (8192/16) * (512/64) = 512*8; 8 waves / block.
__global__ __launch_bounds__(256) void out_gemm(const float* __restrict__ A,
                                                const float* __restrict__ W,
                                                const float* __restrict__ bias,
                                                float* __restrict__ out) {
  const int lane = threadIdx.x & 31;
  const int wid  = (blockIdx.x * blockDim.x + threadIdx.x) >> 5;
  const int mt   = wid >> 3;
  const int ns   = wid & 7;
  const int m    = lane & 15;
  const int hl   = lane >> 4;
  const int col0 = ns * 64;
  const float* arow = A + (size_t)(mt * 16 + m) * Dn;

  v8f acc[4] = {v8f{}, v8f{}, v8f{}, v8f{}};
#pragma unroll 4
  for (int ks = 0; ks < Dn / 4; ++ks) {
    const int kb = ks * 4 + hl * 2;
    v2f a = *(const v2f*)(arow + kb);
    const float* w0 = W + (size_t)kb * Dn + col0 + m;
    const float* w1 = w0 + Dn;
#pragma unroll
    for (int j = 0; j < 4; ++j) {
      v2f b;
      b.x = w0[16 * j];
      b.y = w1[16 * j];
      acc[j] = wmma4(a, b, acc[j]);
    }
  }
#pragma unroll
  for (int j = 0; j < 4; ++j) {
    const float bb = bias[col0 + j * 16 + m];
#pragma unroll
    for (int r = 0; r < 8; ++r) {
      const int grow = mt * 16 + r + hl * 8;
      out[(size_t)grow * Dn + col0 + j * 16 + m] = acc[j][r] + bb;
    }
  }
}

}  // namespace

extern "C" void kernel_launch(void* const* d_in, const int* in_sizes, int n_in,
                              void* d_out, int out_size, void* d_ws,
                              size_t ws_size, hipStream_t stream) {
  (void)in_sizes; (void)n_in; (void)out_size; (void)ws_size;
  const float* x     = (const float*)d_in[0];
  const int*   mask  = (const int*)d_in[1];
  const float* W_in  = (const float*)d_in[2];
  const float* b_in  = (const float*)d_in[3];
  const float* W_out = (const float*)d_in[4];
  const float* b_out = (const float*)d_in[5];
  const float* pos_K = (const float*)d_in[6];
  const float* pos_V = (const float*)d_in[7];
  float* out = (float*)d_out;

  const size_t headElems = (size_t)Bn * Hn * Sn * DHn;  // 4 Mi floats
  float* Q  = (float*)d_ws;
  float* K  = Q + headElems;
  float* V  = K + headElems;
  float* AT = V + headElems;                            // [B,S,D]

  // K1: 512 Mtiles * 24 strips = 12288 waves, 8 waves/block
  qkv_gemm<<<dim3(12288 / 8), dim3(256), 0, stream>>>(x, W_in, b_in, Q, K, V);
  // K2: one wave per (b,h,qblock)
  attn<<<dim3(Bn * Hn * (Sn / 16)), dim3(32), 0, stream>>>(Q, K, V, mask,
                                                           pos_K, pos_V, AT);
  // K3: 512 Mtiles * 8 strips = 4096 waves, 8 waves/block
  out_gemm<<<dim3(4096 / 8), dim3(256), 0, stream>>>(AT, W_out, b_out, out);
}